// GNNWrapper_50714973831901
// MI455X (gfx1250) — compile-verified
//
#include <hip/hip_runtime.h>
#include <hip/hip_bf16.h>
#include <math.h>

// Problem constants (from reference): B=1024, N=256, F=80, C=16, H=256
#define BB 1024
#define NN 256
#define FF 80
#define CC 16
#define HH 256
#define KD (NN*CC)   // 4096, dense K dim

typedef float v2f __attribute__((ext_vector_type(2)));
typedef float v8f __attribute__((ext_vector_type(8)));

__device__ __forceinline__ v8f wmma4(v2f a, v2f b, v8f c) {
    // D = A(16x4 f32) * B(4x16 f32) + C(16x16 f32)
    return __builtin_amdgcn_wmma_f32_16x16x4_f32(false, a, false, b, (short)0, c,
                                                 false, false);
}

// -------- attention phase: e-vectors, per-row softmax stats, out = P @ Hf + bias
__device__ void attention_phase(const float* sHf, float* sOut,
                                const unsigned* sAbits,
                                float* sEs, float* sEn, float* sMx, float* sRw,
                                const float* sAv, const float* sBias,
                                int tid, int lane, int wave)
{
    // ---- e_self / e_neigh (one node per thread) ----
    {
        float es = 0.f, en = 0.f;
        #pragma unroll
        for (int c = 0; c < CC; ++c) {
            float h = sHf[tid * CC + c];
            es += h * sAv[c];
            en += h * sAv[CC + c];
        }
        sEs[tid] = es;
        sEn[tid] = en;
    }
    __syncthreads();

    // ---- per-row softmax statistics (max, 1/sum) ----
    {
        const float esr = sEs[tid];
        float mx = -3.402823466e38f;
        for (int jc = 0; jc < 8; ++jc) {
            unsigned bits = sAbits[tid * 8 + jc];
            #pragma unroll
            for (int jj = 0; jj < 32; ++jj) {
                float raw = esr + sEn[jc * 32 + jj];
                float lr  = raw > 0.f ? raw : 0.2f * raw;        // leaky_relu 0.2
                float sc  = ((bits >> jj) & 1u) ? lr : lr - 1e10f; // MASK=-10e9
                mx = fmaxf(mx, sc);
            }
        }
        float s = 0.f;
        for (int jc = 0; jc < 8; ++jc) {
            unsigned bits = sAbits[tid * 8 + jc];
            #pragma unroll
            for (int jj = 0; jj < 32; ++jj) {
                float raw = esr + sEn[jc * 32 + jj];
                float lr  = raw > 0.f ? raw : 0.2f * raw;
                float sc  = ((bits >> jj) & 1u) ? lr : lr - 1e10f;
                s += __expf(sc - mx);                 // masked -> exp(~-1e10)=0
            }
        }
        sMx[tid] = mx;
        sRw[tid] = 1.f / s;
    }
    __syncthreads();

    // ---- out = P @ Hf via V_WMMA_F32_16X16X4_F32; P computed on the fly ----
    {
        const int   n    = lane & 15;                 // output column
        const int   rsel = lane & 15;                 // A-matrix row in tile
        const int   koff = (lane >= 16) ? 2 : 0;      // A/B K-pair per half-wave
        const float bias = sBias[n];

        for (int t = 0; t < 2; ++t) {                 // 8 waves x 2 = 16 M-tiles
            const int mt  = wave + t * 8;
            const int row = mt * 16 + rsel;
            const float esr = sEs[row];
            const float mr  = sMx[row];
            const float rr  = sRw[row];
            v8f acc = {};
            for (int kk = 0; kk < 64; ++kk) {         // K = 256, steps of 4
                const int k0 = kk * 4 + koff;
                const int k1 = k0 + 1;
                // attn(row, k0), attn(row, k1)
                unsigned bits0 = sAbits[row * 8 + (k0 >> 5)];
                float raw0 = esr + sEn[k0];
                float lr0  = raw0 > 0.f ? raw0 : 0.2f * raw0;
                float sc0  = ((bits0 >> (k0 & 31)) & 1u) ? lr0 : lr0 - 1e10f;
                float p0   = __expf(sc0 - mr) * rr;

                unsigned bits1 = sAbits[row * 8 + (k1 >> 5)];
                float raw1 = esr + sEn[k1];
                float lr1  = raw1 > 0.f ? raw1 : 0.2f * raw1;
                float sc1  = ((bits1 >> (k1 & 31)) & 1u) ? lr1 : lr1 - 1e10f;
                float p1   = __expf(sc1 - mr) * rr;

                v2f a; a.x = p0; a.y = p1;
                v2f b; b.x = sHf[k0 * CC + n]; b.y = sHf[k1 * CC + n];
                acc = wmma4(a, b, acc);
            }
            #pragma unroll
            for (int r = 0; r < 8; ++r) {
                int mrow = mt * 16 + r + ((lane >= 16) ? 8 : 0);
                sOut[mrow * CC + n] = acc[r] + bias;
            }
        }
    }
    __syncthreads();
}

// -------- fused 2-layer GAT, one workgroup (256 thr / 8 waves) per batch elem
__global__ __launch_bounds__(256) void gat_fused(
    const float* __restrict__ X,  const float* __restrict__ A,
    const float* __restrict__ W1, const float* __restrict__ a1s,
    const float* __restrict__ a1n,const float* __restrict__ b1,
    const float* __restrict__ W2, const float* __restrict__ a2s,
    const float* __restrict__ a2n,const float* __restrict__ b2,
    float* __restrict__ X2out)
{
    __shared__ float    sHf[NN * CC];      // current layer features (GEMM B operand)
    __shared__ float    sOut[NN * CC];     // aggregated output
    __shared__ unsigned sAbits[NN * 8];    // 256x256 adjacency as bits (8 KB)
    __shared__ float    sEs[NN], sEn[NN], sMx[NN], sRw[NN];
    __shared__ float    sW[FF * CC];       // layer weights
    __shared__ float    sAv[2 * CC];       // a_self, a_neigh
    __shared__ float    sBias[CC];

    const int tid  = threadIdx.x;
    const int lane = tid & 31;
    const int wave = tid >> 5;
    const int b    = blockIdx.x;

    // ---- 1. compress A[b] into bitmask via wave32 ballot (A read ONCE) ----
    {
        const float* Ab = A + (size_t)b * NN * NN;
        for (int t = 0; t < 256; ++t) {
            int widx  = wave * 256 + t;          // 2048 words / 8 waves
            int row   = widx >> 3;
            int chunk = widx & 7;
            float av  = Ab[(size_t)row * NN + chunk * 32 + lane];
            unsigned m = (unsigned)__ballot(av != 0.0f);
            if (lane == 0) sAbits[widx] = m;
        }
    }

    // ---- 2. stage layer-1 params ----
    for (int i = tid; i < FF * CC; i += 256) sW[i] = W1[i];
    if (tid < CC) { sAv[tid] = a1s[tid]; sAv[CC + tid] = a1n[tid]; sBias[tid] = b1[tid]; }
    __syncthreads();

    // ---- 3. Hf1 = X[b] @ W1  (256x80 @ 80x16, one row per thread) ----
    {
        const float4* Xr = (const float4*)(X + (size_t)b * NN * FF + (size_t)tid * FF);
        float acc[CC];
        #pragma unroll
        for (int c = 0; c < CC; ++c) acc[c] = 0.f;
        for (int k4 = 0; k4 < FF / 4; ++k4) {
            float4 xv = Xr[k4];
            const int k = k4 * 4;
            #pragma unroll
            for (int c = 0; c < CC; ++c) {
                acc[c] += xv.x * sW[(k + 0) * CC + c];
                acc[c] += xv.y * sW[(k + 1) * CC + c];
                acc[c] += xv.z * sW[(k + 2) * CC + c];
                acc[c] += xv.w * sW[(k + 3) * CC + c];
            }
        }
        #pragma unroll
        for (int c = 0; c < CC; ++c) sHf[tid * CC + c] = acc[c];
    }
    __syncthreads();

    // ---- 4. layer-1 attention + aggregation ----
    attention_phase(sHf, sOut, sAbits, sEs, sEn, sMx, sRw, sAv, sBias, tid, lane, wave);

    // ---- 5. stage layer-2 params ----
    for (int i = tid; i < CC * CC; i += 256) sW[i] = W2[i];
    if (tid < CC) { sAv[tid] = a2s[tid]; sAv[CC + tid] = a2n[tid]; sBias[tid] = b2[tid]; }
    __syncthreads();

    // ---- 6. Hf2 = X1 @ W2 (256x16 @ 16x16) ----
    {
        float acc[CC];
        #pragma unroll
        for (int c = 0; c < CC; ++c) acc[c] = 0.f;
        #pragma unroll
        for (int k = 0; k < CC; ++k) {
            float xv = sOut[tid * CC + k];
            #pragma unroll
            for (int c = 0; c < CC; ++c) acc[c] += xv * sW[k * CC + c];
        }
        __syncthreads();
        #pragma unroll
        for (int c = 0; c < CC; ++c) sHf[tid * CC + c] = acc[c];
    }
    __syncthreads();

    // ---- 7. layer-2 attention + aggregation (bitmask reused) ----
    attention_phase(sHf, sOut, sAbits, sEs, sEn, sMx, sRw, sAv, sBias, tid, lane, wave);

    // ---- 8. write X2[b] ----
    for (int i = tid; i < NN * CC; i += 256)
        X2out[(size_t)b * NN * CC + i] = sOut[i];
}

// -------- dense head: out = tanh(flat @ Wd + bd), 1024x4096 @ 4096x256
__global__ __launch_bounds__(256) void dense_tanh(
    const float* __restrict__ flat, const float* __restrict__ Wd,
    const float* __restrict__ bd,   float* __restrict__ out)
{
    __shared__ float sA[16 * 132];            // padded: stride 132 avoids bank conflicts

    const int tid  = threadIdx.x;
    const int lane = tid & 31;
    const int wave = tid >> 5;
    const int mt   = blockIdx.x;              // 0..63 : 16-row tile of flat
    const int nt   = blockIdx.y * 8 + wave;   // 0..15 : 16-col tile of out
    const int n    = lane & 15;
    const int col  = nt * 16 + n;
    const int rsel = lane & 15;
    const int koff = (lane >= 16) ? 2 : 0;

    v8f acc = {};
    for (int kc = 0; kc < KD; kc += 128) {
        __syncthreads();
        for (int i = tid; i < 16 * 128; i += 256) {
            int r = i >> 7, k = i & 127;
            sA[r * 132 + k] = flat[(size_t)(mt * 16 + r) * KD + kc + k];
        }
        __syncthreads();
        for (int kk = 0; kk < 32; ++kk) {
            const int k0 = kk * 4 + koff;
            v2f a; a.x = sA[rsel * 132 + k0]; a.y = sA[rsel * 132 + k0 + 1];
            v2f b;
            b.x = Wd[(size_t)(kc + k0)     * HH + col];
            b.y = Wd[(size_t)(kc + k0 + 1) * HH + col];
            acc = wmma4(a, b, acc);
        }
    }
    const float bias = bd[col];
    #pragma unroll
    for (int r = 0; r < 8; ++r) {
        int m = mt * 16 + r + ((lane >= 16) ? 8 : 0);
        out[(size_t)m * HH + col] = tanhf(acc[r] + bias);
    }
}

extern "C" void kernel_launch(void* const* d_in, const int* in_sizes, int n_in,
                              void* d_out, int out_size, void* d_ws, size_t ws_size,
                              hipStream_t stream) {
    (void)in_sizes; (void)n_in; (void)out_size; (void)ws_size;
    const float* X   = (const float*)d_in[0];
    const float* A   = (const float*)d_in[1];
    const float* W1  = (const float*)d_in[2];
    const float* a1s = (const float*)d_in[3];
    const float* a1n = (const float*)d_in[4];
    const float* b1  = (const float*)d_in[5];
    const float* W2  = (const float*)d_in[6];
    const float* a2s = (const float*)d_in[7];
    const float* a2n = (const float*)d_in[8];
    const float* b2  = (const float*)d_in[9];
    const float* Wd  = (const float*)d_in[10];
    const float* bd  = (const float*)d_in[11];
    float* out = (float*)d_out;
    float* X2  = (float*)d_ws;                 // 1024*4096 f32 = 16 MB scratch

    gat_fused<<<dim3(BB), dim3(256), 0, stream>>>(X, A, W1, a1s, a1n, b1,
                                                  W2, a2s, a2n, b2, X2);
    dense_tanh<<<dim3(64, 2), dim3(256), 0, stream>>>(X2, Wd, bd, out);
}